// KVCache_57595511439470
// MI455X (gfx1250) — compile-verified
//
#include <hip/hip_runtime.h>
#include <hip/hip_bf16.h>

// KV-cache update + GQA expansion for MI455X (gfx1250).
// Zero-FLOP, pure-bandwidth op: ~2.5 GB of traffic -> ~107us at 23.3 TB/s.
// Strategy: 32-bit indexed (saddr + scale_offset voffset) b128 non-temporal
// streaming with 8-deep load clauses, fully unrolled compile-time trip
// counts, and static global_prefetch_b8 lookahead (8 iterations ahead).
// Cache read once; n_rep head expansion broadcast-written from registers.

typedef __attribute__((ext_vector_type(4))) unsigned int v4u; // 16B = 8 bf16

namespace {
constexpr unsigned L = 16, B = 4, S = 4096, Hkv = 8, D = 128;
constexpr unsigned N_REP = 4, Hq = Hkv * N_REP;                  // 32
constexpr unsigned CACHE_CHUNKS = L * B * S * Hkv * D / 8;       // 2^25
constexpr unsigned KEYS_CHUNKS  = B * S * Hq * D / 8;            // 2^23
constexpr unsigned SRC_CHUNKS   = B * S * Hkv * D / 8;           // 2^21

constexpr unsigned COPY_BLOCKS  = 4096;
constexpr unsigned COPY_THREADS = COPY_BLOCKS * 256u;            // 2^20
constexpr int      COPY_ITERS   = (int)(CACHE_CHUNKS / COPY_THREADS); // 32
constexpr int      GROUP        = 4;                             // chunk-pairs/group
constexpr int      NGROUPS      = COPY_ITERS / GROUP;            // 8
constexpr int      PF_ITERS     = 8;                             // lookahead (iters)

constexpr unsigned EXP_BLOCKS   = 2048;
constexpr unsigned EXP_THREADS  = EXP_BLOCKS * 256u;             // 2^19
constexpr int      EXP_ITERS    = (int)(SRC_CHUNKS / EXP_THREADS);    // 4 per tensor
} // namespace

// --- Bulk copy: k -> new_k, v -> new_v (dominant 2.15 GB of traffic) ---
// Fully unrolled: 8 groups x (4 k-chunks + 4 v-chunks). Loads are batched
// into register arrays so the backend emits one 8-deep s_clause of
// global_load_b128 th:NT before the first dependent store (4 KB in flight
// per wave). Prefetches are statically guarded: every chunk in iterations
// [8,32) is prefetched into GL2 exactly once, 8 iterations early.
__global__ __launch_bounds__(256) void kv_copy_kernel(
    const v4u* __restrict__ k, const v4u* __restrict__ v,
    v4u* __restrict__ nk, v4u* __restrict__ nv) {
  const unsigned tid = blockIdx.x * 256u + threadIdx.x;          // < 2^20
#pragma unroll
  for (int g = 0; g < NGROUPS; ++g) {
    const unsigned base = tid + (unsigned)(g * GROUP) * COPY_THREADS;
    if (g * GROUP + PF_ITERS < COPY_ITERS) {    // static after full unroll
#pragma unroll
      for (int r = 0; r < GROUP; ++r) {
        const unsigned p = base + (unsigned)(r + PF_ITERS) * COPY_THREADS;
        __builtin_prefetch(k + p, 0, 0);        // global_prefetch_b8 (GL2)
        __builtin_prefetch(v + p, 0, 0);
      }
    }
    v4u a[GROUP], b[GROUP];
#pragma unroll
    for (int r = 0; r < GROUP; ++r) {
      const unsigned i = base + (unsigned)r * COPY_THREADS;
      a[r] = __builtin_nontemporal_load(k + i); // global_load_b128 th:NT
      b[r] = __builtin_nontemporal_load(v + i);
    }
#pragma unroll
    for (int r = 0; r < GROUP; ++r) {
      const unsigned i = base + (unsigned)r * COPY_THREADS;
      __builtin_nontemporal_store(a[r], nk + i); // global_store_b128 th:NT
      __builtin_nontemporal_store(b[r], nv + i);
    }
  }
}

// --- Fixup: write xk/xv into new_k/new_v at [layer_idx, :, cur_pos, :, :] ---
// Insert region is B*Hkv*D = 4096 elements (512 chunks) per tensor.
__global__ __launch_bounds__(256) void kv_fixup_kernel(
    const v4u* __restrict__ xk, const v4u* __restrict__ xv,
    v4u* __restrict__ nk, v4u* __restrict__ nv,
    const int* __restrict__ layer_p, const int* __restrict__ pos_p) {
  const unsigned layer = (unsigned)*layer_p;
  const unsigned pos   = (unsigned)*pos_p;
  const unsigned tid = blockIdx.x * 256u + threadIdx.x;          // 0..1023
  const unsigned c   = tid & 511u;       // chunk within [B][Hkv*D/8]
  const unsigned b   = c >> 7;           // Hkv*D/8 = 128 chunks per batch
  const unsigned r   = c & 127u;
  const unsigned dst = ((layer * B + b) * S + pos) * 128u + r;
  if (tid < 512u) nk[dst] = xk[c];
  else            nv[dst] = xv[c];
}

// --- Expansion: keys/values[b,s,hq,:] = updated_cache[layer,b,s,hq/4,:] ---
// Each source chunk is read ONCE and stored N_REP times (dest chunks are
// D elements = 16 chunks apart, coalesced across the wave). The inserted
// token (s == cur_pos) is sourced straight from xk/xv, so this kernel never
// re-reads d_out and has no dependency on the copy kernel. The k/v halves
// are statically selected (no per-lane pointer selects).
__global__ __launch_bounds__(256) void kv_expand_kernel(
    const v4u* __restrict__ k, const v4u* __restrict__ v,
    const v4u* __restrict__ xk, const v4u* __restrict__ xv,
    v4u* __restrict__ keys, v4u* __restrict__ vals,
    const int* __restrict__ layer_p, const int* __restrict__ pos_p) {
  const unsigned layer = (unsigned)*layer_p;
  const unsigned pos   = (unsigned)*pos_p;
  const unsigned tid = blockIdx.x * 256u + threadIdx.x;          // < 2^19
#pragma unroll
  for (int half = 0; half < 2; ++half) {
    const v4u* __restrict__ cache = half ? v : k;
    const v4u* __restrict__ xnew  = half ? xv : xk;
    v4u* __restrict__ dst         = half ? vals : keys;
#pragma unroll
    for (int j = 0; j < EXP_ITERS; ++j) {
      const unsigned c   = tid + (unsigned)j * EXP_THREADS;      // < 2^21
      // Decompose c: [b][s][hkv][dchunk], all power-of-two extents.
      const unsigned b   = c >> 19;                // S*Hkv*D/8 = 2^19 per batch
      const unsigned rem = c & ((1u << 19) - 1u);
      const unsigned s   = rem >> 7;               // 128 chunks per (b,s)
      const unsigned q   = rem & 127u;             // = hkv*16 + dchunk
      const v4u data = (s == pos)
          ? xnew[(b << 7) + q]                                   // fresh token
          : cache[(((layer << 2) + b) * S + s) * 128u + q];      // cache
      const unsigned hkv  = q >> 4;
      const unsigned dc   = q & 15u;
      const unsigned base = (((b << 12) + s) * Hq + (hkv << 2)) * 16u + dc;
#pragma unroll
      for (unsigned rr = 0; rr < N_REP; ++rr)
        __builtin_nontemporal_store(data, dst + base + rr * 16u);
    }
  }
}

extern "C" void kernel_launch(void* const* d_in, const int* in_sizes, int n_in,
                              void* d_out, int out_size, void* d_ws, size_t ws_size,
                              hipStream_t stream) {
  const v4u* k  = (const v4u*)d_in[0];  // [L,B,S,Hkv,D] bf16
  const v4u* v  = (const v4u*)d_in[1];
  const v4u* xk = (const v4u*)d_in[2];  // [B,1,Hkv,D] bf16
  const v4u* xv = (const v4u*)d_in[3];
  const int* layer_p = (const int*)d_in[4];
  const int* pos_p   = (const int*)d_in[5];
  // d_in[6] = n_rep; fixed at 4 by the output geometry the harness allocated.

  // d_out layout (flat, return order): keys, values, new_k, new_v
  v4u* keys = (v4u*)d_out;
  v4u* vals = keys + KEYS_CHUNKS;
  v4u* nk   = vals + KEYS_CHUNKS;
  v4u* nv   = nk + CACHE_CHUNKS;

  // 1) bulk cache copy: exactly 32 chunks per thread, no bounds checks.
  kv_copy_kernel<<<COPY_BLOCKS, 256, 0, stream>>>(k, v, nk, nv);
  // 2) overwrite the inserted token slice (ordered after copy; same stream).
  kv_fixup_kernel<<<4, 256, 0, stream>>>(xk, xv, nk, nv, layer_p, pos_p);
  // 3) GQA head expansion into keys/values (independent of 1/2).
  kv_expand_kernel<<<EXP_BLOCKS, 256, 0, stream>>>(k, v, xk, xv, keys, vals,
                                                   layer_p, pos_p);
}